// LRF_70695161692718
// MI455X (gfx1250) — compile-verified
//
#include <hip/hip_runtime.h>
#include <math.h>

typedef __attribute__((ext_vector_type(2))) float v2f;
typedef __attribute__((ext_vector_type(8))) float v8f;

__device__ __forceinline__ float wave_sum32(float v) {
#pragma unroll
  for (int m = 16; m > 0; m >>= 1) v += __shfl_xor(v, m, 32);
  return v;
}

// One wave32 per point. Layout of inputs (row-major):
//   xyz:  [B][3][1], xyz_group: [B][3][64], r_lrf: [B], out: [B][3][64]
__global__ __launch_bounds__(256)
void lrf_kernel(const float* __restrict__ xyz,
                const float* __restrict__ xyzg,
                const float* __restrict__ rlrf,
                float* __restrict__ out,
                int B)
{
  const int lane = threadIdx.x & 31;
  const int w    = (int)((blockIdx.x * blockDim.x + threadIdx.x) >> 5);
  if (w >= B) return;  // wave-uniform: EXEC stays all-ones for live waves (WMMA req.)

  const float cx = xyz[3 * w + 0];
  const float cy = xyz[3 * w + 1];
  const float cz = xyz[3 * w + 2];
  const float r  = rlrf[w];
  const float rinv = 1.0f / r;

  // g = xyz_group - xyz  (== -x_initial == the "x" after negation in the reference)
  // lane owns columns {lane, lane+32}; each row load is a contiguous 128B transaction.
  const float* gp = xyzg + (size_t)w * 192;
  const float g0a = gp[      lane]      - cx;
  const float g1a = gp[ 64 + lane]      - cy;
  const float g2a = gp[128 + lane]      - cz;
  const float g0b = gp[      lane + 32] - cx;
  const float g1b = gp[ 64 + lane + 32] - cy;
  const float g2b = gp[128 + lane + 32] - cz;

  // ---- covariance xxt = g g^T / 64 (sign cancels vs x_initial) ----
  float s00 = g0a * g0a + g0b * g0b;
  float s01 = g0a * g1a + g0b * g1b;
  float s02 = g0a * g2a + g0b * g2b;
  float s11 = g1a * g1a + g1b * g1b;
  float s12 = g1a * g2a + g1b * g2b;
  float s22 = g2a * g2a + g2b * g2b;
  const float a00 = wave_sum32(s00) * (1.0f / 64.0f);
  const float a01 = wave_sum32(s01) * (1.0f / 64.0f);
  const float a02 = wave_sum32(s02) * (1.0f / 64.0f);
  const float a11 = wave_sum32(s11) * (1.0f / 64.0f);
  const float a12 = wave_sum32(s12) * (1.0f / 64.0f);
  const float a22 = wave_sum32(s22) * (1.0f / 64.0f);

  // ---- smallest-eigenvalue eigenvector of symmetric 3x3 (analytic, per-lane) ----
  const float p1 = a01 * a01 + a02 * a02 + a12 * a12;
  const float q  = (a00 + a11 + a22) * (1.0f / 3.0f);
  const float b00 = a00 - q, b11 = a11 - q, b22 = a22 - q;
  const float p2 = b00 * b00 + b11 * b11 + b22 * b22 + 2.0f * p1;
  const float p  = sqrtf(p2 * (1.0f / 6.0f)) + 1e-20f;
  const float pi_ = 1.0f / p;
  const float c00 = b00 * pi_, c01 = a01 * pi_, c02 = a02 * pi_;
  const float c11 = b11 * pi_, c12 = a12 * pi_, c22 = b22 * pi_;
  float detB = c00 * (c11 * c22 - c12 * c12)
             - c01 * (c01 * c22 - c12 * c02)
             + c02 * (c01 * c12 - c11 * c02);
  float rr = 0.5f * detB;
  rr = fminf(1.0f, fmaxf(-1.0f, rr));
  const float phi = acosf(rr) * (1.0f / 3.0f);
  const float lam = q + 2.0f * p * cosf(phi + 2.0943951023931953f);  // smallest eig

  // eigenvector via largest cross product of rows of (A - lam*I)
  const float r0x = a00 - lam, r0y = a01,       r0z = a02;
  const float r1x = a01,       r1y = a11 - lam, r1z = a12;
  const float r2x = a02,       r2y = a12,       r2z = a22 - lam;
  float c1x = r0y * r1z - r0z * r1y, c1y = r0z * r1x - r0x * r1z, c1z = r0x * r1y - r0y * r1x;
  float c2x = r0y * r2z - r0z * r2y, c2y = r0z * r2x - r0x * r2z, c2z = r0x * r2y - r0y * r2x;
  float c3x = r1y * r2z - r1z * r2y, c3y = r1z * r2x - r1x * r2z, c3z = r1x * r2y - r1y * r2x;
  const float n1 = c1x * c1x + c1y * c1y + c1z * c1z;
  const float n2 = c2x * c2x + c2y * c2y + c2z * c2z;
  const float n3 = c3x * c3x + c3y * c3y + c3z * c3z;
  float vx = c1x, vy = c1y, vz = c1z, nn = n1;
  if (n2 > nn) { vx = c2x; vy = c2y; vz = c2z; nn = n2; }
  if (n3 > nn) { vx = c3x; vy = c3y; vz = c3z; nn = n3; }
  if (nn < 1e-30f) { vx = 0.0f; vy = 0.0f; vz = 1.0f; nn = 1.0f; }
  const float inl = rsqrtf(nn);
  const float v0 = vx * inl, v1 = vy * inl, v2 = vz * inl;

  // ---- sign count + weighted vi accumulation (vi_c is sign-independent) ----
  const float dva = v0 * g0a + v1 * g1a + v2 * g2a;   // v . g (column a)
  const float dvb = v0 * g0b + v1 * g1b + v2 * g2b;
  // center_proj = v . x_initial = -(v . g)
  float cnt = 0.0f;
  cnt += (-dva >  0.001f) ? 1.0f : 0.0f;
  cnt -= (-dva < -0.001f) ? 1.0f : 0.0f;
  cnt += (-dvb >  0.001f) ? 1.0f : 0.0f;
  cnt -= (-dvb < -0.001f) ? 1.0f : 0.0f;

  const float la = sqrtf(g0a * g0a + g1a * g1a + g2a * g2a);
  const float lb = sqrtf(g0b * g0b + g1b * g1b + g2b * g2b);
  float ta = r - la;  const float wa = ta * ta * dva * dva;   // alpha * beta
  float tb = r - lb;  const float wb = tb * tb * dvb * dvb;

  float ax = wa * (g0a - v0 * dva) + wb * (g0b - v0 * dvb);
  float ay = wa * (g1a - v1 * dva) + wb * (g1b - v1 * dvb);
  float az = wa * (g2a - v2 * dva) + wb * (g2b - v2 * dvb);

  cnt = wave_sum32(cnt);
  ax  = wave_sum32(ax);
  ay  = wave_sum32(ay);
  az  = wave_sum32(az);

  const float sgn = (cnt < 0.0f) ? -1.0f : 1.0f;
  const float zx = sgn * v0, zy = sgn * v1, zz = sgn * v2;
  const float nv = 1.0f / (sqrtf(ax * ax + ay * ay + az * az) + 1e-10f);
  const float xpx = ax * nv, xpy = ay * nv, xpz = az * nv;
  const float ypx = xpy * zz - xpz * zy;
  const float ypy = xpz * zx - xpx * zz;
  const float ypz = xpx * zy - xpy * zx;

  // ---- out(3x64) = (1/r) * [xp;yp;zp] . g  via V_WMMA_F32_16X16X4_F32 ----
  // A (16x4, M=lane%16): lanes 0-15 hold K=0,1; lanes 16-31 hold K=2,3(=0 pad).
  const int m = lane & 15;
  float e0 = 0.0f, e1 = 0.0f, e2 = 0.0f;
  if      (m == 0) { e0 = xpx; e1 = xpy; e2 = xpz; }
  else if (m == 1) { e0 = ypx; e1 = ypy; e2 = ypz; }
  else if (m == 2) { e0 = zx;  e1 = zy;  e2 = zz;  }
  e0 *= rinv; e1 *= rinv; e2 *= rinv;
  v2f A;
  A.x = (lane < 16) ? e0 : e2;
  A.y = (lane < 16) ? e1 : 0.0f;

  float* op = out + (size_t)w * 192;
#pragma unroll
  for (int k = 0; k < 4; ++k) {
    // B (4x16): column chunk [k*16, k*16+16). Source column for this lane's slot:
    const int src = ((k & 1) << 4) | (lane & 15);
    const float q0 = (k < 2) ? g0a : g0b;
    const float q1 = (k < 2) ? g1a : g1b;
    const float q2 = (k < 2) ? g2a : g2b;
    const float t0 = __shfl(q0, src, 32);
    const float t1 = __shfl(q1, src, 32);
    const float t2 = __shfl(q2, src, 32);
    v2f Bm;
    Bm.x = (lane < 16) ? t0 : t2;   // rows K=0 (lo) / K=2 (hi)
    Bm.y = (lane < 16) ? t1 : 0.0f; // rows K=1 (lo) / K=3 pad (hi)
    v8f C = {};
    C = __builtin_amdgcn_wmma_f32_16x16x4_f32(
        /*neg_a=*/false, A, /*neg_b=*/false, Bm,
        /*c_mod=*/(short)0, C, /*reuse_a=*/false, /*reuse_b=*/false);
    // D rows 0..2 live in VGPRs 0..2 on lanes 0..15 (N = lane)
    if (lane < 16) {
      const int col = k * 16 + lane;
      op[       col] = C[0];
      op[ 64 +  col] = C[1];
      op[128 +  col] = C[2];
    }
  }
}

extern "C" void kernel_launch(void* const* d_in, const int* in_sizes, int n_in,
                              void* d_out, int out_size, void* d_ws, size_t ws_size,
                              hipStream_t stream) {
  const float* xyz  = (const float*)d_in[0];   // B*3
  const float* xyzg = (const float*)d_in[1];   // B*3*64
  const float* rlrf = (const float*)d_in[2];   // B
  float* out = (float*)d_out;                  // B*3*64
  const int B = in_sizes[2];
  const int waves_per_block = 256 / 32;
  const int grid = (B + waves_per_block - 1) / waves_per_block;
  lrf_kernel<<<grid, 256, 0, stream>>>(xyz, xyzg, rlrf, out, B);
}